// TraditionalSAE_2027224564261
// MI455X (gfx1250) — compile-verified
//
#include <hip/hip_runtime.h>

typedef float v2f __attribute__((ext_vector_type(2)));
typedef float v8f __attribute__((ext_vector_type(8)));

#define TOK   8192          // B*H*W tokens
#define CIN   1280          // C
#define HID   20480         // hidden
#define KSP   32            // top-k
#define BHW   256           // H*W
#define XBSTRIDE (CIN*BHW)  // per-batch stride in x / recon (327680)

// ---------------- Kernel 1: encoder GEMM + bias + ReLU (f32 WMMA) ----------
// Tile: 64(M) x 128(N), K-chunk 16. 8 waves: wave = (wm in 0..3) x (wn in 0..1),
// each wave owns 16(M) x 64(N) = 4 WMMA n-subtiles.
#define TM 64
#define TN 128
#define TK 16
#define LDA 65
#define LDB 129

__global__ __launch_bounds__(256) void sae_encoder_wmma(
    const float* __restrict__ x, const float* __restrict__ Wenc,
    const float* __restrict__ benc, float* __restrict__ enc)
{
  __shared__ float sA[TK * LDA];   // sA[k][m], 64 wide
  __shared__ float sB[TK * LDB];   // sB[k][n], 128 wide

  const int tid  = threadIdx.x;
  const int lane = tid & 31;
  const int wave = tid >> 5;
  const int wm   = wave & 3;       // M sub-tile (16 rows each)
  const int wn   = wave >> 2;      // N half (64 cols each)
  const int tileM0 = blockIdx.y * TM;
  const int tileN0 = blockIdx.x * TN;
  const int g      = tileM0 >> 8;      // batch index group (64-token tile never crosses)
  const int wbase  = tileM0 & 255;     // h*W+w base
  const int l15    = lane & 15;
  const int koff   = (lane >> 4) * 2;  // K-pair owned by this lane half

  v8f acc[4];
  acc[0] = {}; acc[1] = {}; acc[2] = {}; acc[3] = {};

  // A-tile loader coords: thread t loads A[m4..m4+3][k] (contiguous in memory along m)
  const int lk  = tid >> 4;            // 0..15
  const int lm4 = (tid & 15) * 4;      // 0..60
  // B-tile loader coords: thread t loads W_enc[n][k0+kh .. +7] (contiguous along k)
  const int ln  = tid & 127;           // 0..127
  const int lkh = (tid >> 7) * 8;      // 0 or 8

  for (int k0 = 0; k0 < CIN; k0 += TK) {
    // ---- stage A: x_flat[token, c] = x[g*XBSTRIDE + c*BHW + (wbase+m)]
    {
      const float4 a4 = *(const float4*)(x + (size_t)g * XBSTRIDE
                                           + (size_t)(k0 + lk) * BHW + wbase + lm4);
      sA[lk * LDA + lm4 + 0] = a4.x;
      sA[lk * LDA + lm4 + 1] = a4.y;
      sA[lk * LDA + lm4 + 2] = a4.z;
      sA[lk * LDA + lm4 + 3] = a4.w;
    }
    // ---- stage B: W_enc[hid, c] row-major, contiguous in c
    {
      const float4* src = (const float4*)(Wenc + (size_t)(tileN0 + ln) * CIN + k0 + lkh);
      const float4 b0 = src[0];
      const float4 b1 = src[1];
      sB[(lkh + 0) * LDB + ln] = b0.x;
      sB[(lkh + 1) * LDB + ln] = b0.y;
      sB[(lkh + 2) * LDB + ln] = b0.z;
      sB[(lkh + 3) * LDB + ln] = b0.w;
      sB[(lkh + 4) * LDB + ln] = b1.x;
      sB[(lkh + 5) * LDB + ln] = b1.y;
      sB[(lkh + 6) * LDB + ln] = b1.z;
      sB[(lkh + 7) * LDB + ln] = b1.w;
    }
    __syncthreads();

#pragma unroll
    for (int ks = 0; ks < TK; ks += 4) {
      v2f a;
      a.x = sA[(ks + koff)     * LDA + wm * 16 + l15];
      a.y = sA[(ks + koff + 1) * LDA + wm * 16 + l15];
#pragma unroll
      for (int nt = 0; nt < 4; ++nt) {
        const int nn = wn * 64 + nt * 16 + l15;
        v2f b;
        b.x = sB[(ks + koff)     * LDB + nn];
        b.y = sB[(ks + koff + 1) * LDB + nn];
        acc[nt] = __builtin_amdgcn_wmma_f32_16x16x4_f32(
            false, a, false, b, (short)0, acc[nt], false, false);
      }
    }
    __syncthreads();
  }

  // ---- epilogue: bias + ReLU, store enc[token, hid]
  const int mrow_off = 8 * (lane >> 4);
#pragma unroll
  for (int nt = 0; nt < 4; ++nt) {
    const int hid = tileN0 + wn * 64 + nt * 16 + l15;
    const float bias = benc[hid];
#pragma unroll
    for (int j = 0; j < 8; ++j) {
      const int token = tileM0 + wm * 16 + j + mrow_off;
      float v = acc[nt][j] + bias;
      v = v > 0.0f ? v : 0.0f;
      enc[(size_t)token * HID + hid] = v;
    }
  }
}

// ---------------- Kernel 2: per-row top-32 select (in place) + sparse decode
#define NBINS    4096
#define CAND_MAX 2048

__global__ __launch_bounds__(256) void sae_topk_decode(
    float* __restrict__ enc, const float* __restrict__ Wdec,
    const float* __restrict__ bdec, float* __restrict__ recon)
{
  __shared__ int   hist[NBINS];
  __shared__ float candVal[CAND_MAX];
  __shared__ int   candIdx[CAND_MAX];
  __shared__ float selVal[KSP];
  __shared__ int   selIdx[KSP];
  __shared__ float redVal[256];
  __shared__ int   redIdx[256];
  __shared__ int   redPos[256];
  __shared__ int   sT, sAbove, sCandCnt, sKeptCnt;

  const int tid = threadIdx.x;
  const int row = blockIdx.x;
  float* rowp = enc + (size_t)row * HID;

  for (int i = tid; i < NBINS; i += 256) hist[i] = 0;
  if (tid == 0) { sCandCnt = 0; sKeptCnt = 0; }
  __syncthreads();

  // 1) histogram of monotone float keys (all values >= 0 after ReLU)
  for (int i = tid; i < HID; i += 256) {
    const unsigned key = __float_as_uint(rowp[i]) >> 20;  // < 4096
    atomicAdd(&hist[key], 1);
  }
  __syncthreads();

  // 2) threshold bin: first bin (from top) where cumulative count >= K
  if (tid == 0) {
    int cum = 0, T = 0, above = 0;
    for (int b = NBINS - 1; b >= 0; --b) {
      const int h = hist[b];
      if (cum + h >= KSP) { T = b; above = cum; break; }
      cum += h;
    }
    sT = T; sAbove = above;
  }
  __syncthreads();
  const int T = sT;
  const int above = sAbove;   // count strictly above bin T, always < K

  // 3) rescan: keep > T, zero < T, compact == T candidates
  for (int i = tid; i < HID; i += 256) {
    const float v = rowp[i];
    const int key = (int)(__float_as_uint(v) >> 20);
    if (key > T) {
      const int p = atomicAdd(&sKeptCnt, 1);
      if (p < KSP) { selVal[p] = v; selIdx[p] = i; }
    } else if (key == T) {
      const int p = atomicAdd(&sCandCnt, 1);
      if (p < CAND_MAX) { candVal[p] = v; candIdx[p] = i; }
      else rowp[i] = 0.0f;   // overflow only under massive exact ties
    } else {
      rowp[i] = 0.0f;
    }
  }
  __syncthreads();

  // 4) pick remaining R = K - above from candidates (argmax, smaller index on tie)
  int Cn = sCandCnt; if (Cn > CAND_MAX) Cn = CAND_MAX;
  const int R = KSP - above;
  for (int r = 0; r < R; ++r) {
    float bv = -1.0f; int bi = 0x7FFFFFFF; int bp = -1;
    for (int i = tid; i < Cn; i += 256) {
      const float v = candVal[i];
      const int  id = candIdx[i];
      if (v > bv || (v == bv && id < bi)) { bv = v; bi = id; bp = i; }
    }
    redVal[tid] = bv; redIdx[tid] = bi; redPos[tid] = bp;
    __syncthreads();
    for (int s = 128; s > 0; s >>= 1) {
      if (tid < s) {
        const float v2 = redVal[tid + s]; const int i2 = redIdx[tid + s];
        if (v2 > redVal[tid] || (v2 == redVal[tid] && i2 < redIdx[tid])) {
          redVal[tid] = v2; redIdx[tid] = i2; redPos[tid] = redPos[tid + s];
        }
      }
      __syncthreads();
    }
    if (tid == 0) {
      float v = redVal[0]; int id = redIdx[0]; const int p = redPos[0];
      if (p >= 0) candVal[p] = -1.0f;          // remove from pool
      if (v < 0.0f) { v = 0.0f; id = 0; }      // degenerate guard
      selVal[above + r] = v; selIdx[above + r] = id;
    }
    __syncthreads();
  }

  // 5) zero the candidates that were not selected (selected are marked -1)
  for (int i = tid; i < Cn; i += 256) {
    if (candVal[i] >= 0.0f) rowp[candIdx[i]] = 0.0f;
  }
  __syncthreads();

  // 6) sparse decode: out[n, c] = b_dec[c] + sum_k val_k * W_dec[c, idx_k]
  const int b  = row >> 8;
  const int hw = row & 255;
  for (int c = tid; c < CIN; c += 256) {
    const float* wrow = Wdec + (size_t)c * HID;
    float acc = bdec[c];
#pragma unroll
    for (int k = 0; k < KSP; ++k)
      acc += selVal[k] * wrow[selIdx[k]];
    recon[(size_t)b * XBSTRIDE + (size_t)c * BHW + hw] = acc;
  }
}

extern "C" void kernel_launch(void* const* d_in, const int* in_sizes, int n_in,
                              void* d_out, int out_size, void* d_ws, size_t ws_size,
                              hipStream_t stream) {
  (void)in_sizes; (void)n_in; (void)out_size; (void)d_ws; (void)ws_size;
  const float* x    = (const float*)d_in[0];
  const float* Wenc = (const float*)d_in[1];
  const float* benc = (const float*)d_in[2];
  const float* Wdec = (const float*)d_in[3];
  const float* bdec = (const float*)d_in[4];
  float* out   = (float*)d_out;
  float* enc   = out;                       // sparse [TOK, HID]
  float* recon = out + (size_t)TOK * HID;   // [B, C, H, W]

  dim3 grid1(HID / TN, TOK / TM);           // (160, 128)
  sae_encoder_wmma<<<grid1, 256, 0, stream>>>(x, Wenc, benc, enc);
  sae_topk_decode<<<TOK, 256, 0, stream>>>(enc, Wdec, bdec, recon);
}